// _ECELoss_8830452761184
// MI455X (gfx1250) — compile-verified
//
#include <hip/hip_runtime.h>
#include <stdint.h>

// ECE loss for probs[N,128] f32 + labels[N] i32 -> scalar f32.
// Memory-bound (~1.03 GB @ 23.3 TB/s ~= 44 us floor). Strategy:
//   - one wave32 per row, 16B (float4) per lane => 512B coalesced per wave
//   - CDNA5 async global->LDS pipeline, 4-deep ring (global_load_async_to_lds_b128
//     + s_wait_asynccnt): 2KB outstanding per wave => ~32MB in flight machine-wide,
//     enough to cover HBM latency at 23.3 TB/s
//   - hierarchical bin accumulation: LDS float atomics per block, then 48
//     global float atomics per block (tiny atomic traffic, tiny fp error)

#define NBINS 15
#define NCOLS 128
#define ROWS_PER_BLOCK 8   // 8 waves * 1 row each
#define THREADS 256
#define GRID 2048
#define DEPTH 4            // async ring depth (power of 2)

__device__ __forceinline__ void async_load16(unsigned lds_addr, const void* gaddr) {
  // GV mode: 64-bit per-lane address in a VGPR pair, LDS byte address in VDST VGPR.
  asm volatile("global_load_async_to_lds_b128 %0, %1, off"
               :: "v"(lds_addr), "v"((unsigned long long)(uintptr_t)gaddr)
               : "memory");
}

__device__ __forceinline__ void wait_async_pending(int pending) {
  // Retire the oldest outstanding async load: wait ASYNCcnt <= pending-1.
  switch (pending) {
    case 4:  asm volatile("s_wait_asynccnt 0x3" ::: "memory"); break;
    case 3:  asm volatile("s_wait_asynccnt 0x2" ::: "memory"); break;
    case 2:  asm volatile("s_wait_asynccnt 0x1" ::: "memory"); break;
    default: asm volatile("s_wait_asynccnt 0x0" ::: "memory"); break;
  }
}

__global__ void __launch_bounds__(THREADS) ece_partial(
    const float* __restrict__ probs, const int* __restrict__ labels,
    float* __restrict__ gbins, int N) {
  __shared__ __align__(16) float smem[DEPTH][ROWS_PER_BLOCK][NCOLS];  // 16 KB
  __shared__ float sbins[48];  // [0..15)=count, [16..31)=conf_sum, [32..47)=acc_sum

  const int tid  = threadIdx.x;
  const int wave = tid >> 5;
  const int lane = tid & 31;

  if (tid < 48) sbins[tid] = 0.0f;
  __syncthreads();

  const int nchunks = (N + ROWS_PER_BLOCK - 1) / ROWS_PER_BLOCK;
  const int G = (int)gridDim.x;

  // This thread's 16B slot in each ring buffer (low 32 bits of generic ptr = LDS offset).
  unsigned ldsaddr[DEPTH];
  #pragma unroll
  for (int d = 0; d < DEPTH; ++d)
    ldsaddr[d] = (unsigned)(uintptr_t)&smem[d][wave][lane * 4];

  int pending = 0;

  // Prologue: stage up to DEPTH chunks.
  #pragma unroll
  for (int d = 0; d < DEPTH; ++d) {
    const int cc = (int)blockIdx.x + d * G;
    if (cc < nchunks) {
      const int row = cc * ROWS_PER_BLOCK + wave;
      if (row < N) {
        async_load16(ldsaddr[d], probs + (size_t)row * NCOLS + lane * 4);
        ++pending;
      }
    }
  }

  int i = 0;
  for (int c = (int)blockIdx.x; c < nchunks; c += G, ++i) {
    const int buf = i & (DEPTH - 1);
    const int row = c * ROWS_PER_BLOCK + wave;

    if (row < N) {
      // Async loads retire in issue order per wave; the oldest outstanding
      // one is exactly this chunk's load.
      wait_async_pending(pending);
      --pending;

      const float4 v = *reinterpret_cast<const float4*>(&smem[buf][wave][lane * 4]);

      // Per-lane max/argmax over 4 columns, first-index wins on ties.
      float best = v.x; int bi = lane * 4;
      if (v.y > best) { best = v.y; bi = lane * 4 + 1; }
      if (v.z > best) { best = v.z; bi = lane * 4 + 2; }
      if (v.w > best) { best = v.w; bi = lane * 4 + 3; }

      // wave32 reduction; ties -> smaller column index (matches jnp.argmax).
      #pragma unroll
      for (int off = 16; off >= 1; off >>= 1) {
        float ov = __shfl_down(best, off, 32);
        int   oi = __shfl_down(bi,   off, 32);
        if (ov > best || (ov == best && oi < bi)) { best = ov; bi = oi; }
      }

      if (lane == 0) {
        const float conf = best;
        // searchsorted(boundaries, conf, 'left') - 1 == #{j : j/15 < conf} - 1
        int cnt = 0;
        #pragma unroll
        for (int j = 0; j < 16; ++j) cnt += (((float)j / 15.0f) < conf) ? 1 : 0;
        const int bin = cnt - 1;
        if (bin >= 0 && bin < NBINS) {
          const float acc = (bi == labels[row]) ? 1.0f : 0.0f;
          atomicAdd(&sbins[bin],      1.0f);
          atomicAdd(&sbins[16 + bin], conf);
          atomicAdd(&sbins[32 + bin], acc);
        }
      }
    }

    // Refill this buffer with chunk c + DEPTH*G; drain our LDS reads of it first
    // (they were issued this iteration and already consumed, so this wait is effectively free).
    const int nc = c + DEPTH * G;
    if (nc < nchunks) {
      const int nrow = nc * ROWS_PER_BLOCK + wave;
      if (nrow < N) {
        asm volatile("s_wait_dscnt 0x0" ::: "memory");
        async_load16(ldsaddr[buf], probs + (size_t)nrow * NCOLS + lane * 4);
        ++pending;
      }
    }
  }

  __syncthreads();
  if (tid < 48) {
    float p = sbins[tid];
    if (p != 0.0f) atomicAdd(&gbins[tid], p);
  }
}

__global__ void ece_init(float* __restrict__ gbins) {
  if (threadIdx.x < 48) gbins[threadIdx.x] = 0.0f;
}

__global__ void ece_finalize(const float* __restrict__ gbins,
                             float* __restrict__ out, int N) {
  if (threadIdx.x == 0) {
    float ece = 0.0f;
    const float inv_n = 1.0f / (float)N;
    for (int b = 0; b < NBINS; ++b) {
      float cnt = gbins[b];
      if (cnt > 0.0f) {
        float avg_conf = gbins[16 + b] / cnt;
        float avg_acc  = gbins[32 + b] / cnt;
        ece += fabsf(avg_conf - avg_acc) * (cnt * inv_n);
      }
    }
    out[0] = ece;
  }
}

extern "C" void kernel_launch(void* const* d_in, const int* in_sizes, int n_in,
                              void* d_out, int out_size, void* d_ws, size_t ws_size,
                              hipStream_t stream) {
  (void)n_in; (void)out_size; (void)ws_size;
  const float* probs  = (const float*)d_in[0];
  const int*   labels = (const int*)d_in[1];
  const int    N      = in_sizes[1];          // labels flat count == N
  float* gbins = (float*)d_ws;                // 48 floats of scratch

  ece_init<<<1, 64, 0, stream>>>(gbins);
  ece_partial<<<GRID, THREADS, 0, stream>>>(probs, labels, gbins, N);
  ece_finalize<<<1, 32, 0, stream>>>(gbins, (float*)d_out, N);
}